// CrossTransformer_score1_37520834298100
// MI455X (gfx1250) — compile-verified
//
#include <hip/hip_runtime.h>
#include <hip/hip_bf16.h>
#include <math.h>

// ---------------------------------------------------------------------------
// CrossTransformer score on MI455X (gfx1250, wave32).
// All GEMM stages use V_WMMA_F32_16X16X4_F32 (full-FP32 matrix pipe) since
// the Cholesky factorization downstream needs FP32 accuracy.
// ---------------------------------------------------------------------------

typedef __attribute__((ext_vector_type(2))) float v2f;
typedef __attribute__((ext_vector_type(8))) float v8f;

// Problem constants
#define CC   512      // channels
#define DK   128      // projection dim
#define SS   49       // 7*7 spatial
#define NK   25       // b*k*n supports
#define NP   1225     // NK*SS support pixels
#define PPAD 1232     // NP padded to multiple of 16 (and of 4 for K-loop)
#define PW   1280     // support-col stride padded so 16-tiles can over-read
#define QP   245      // 5*49 query pixels
#define QW   320      // query-col stride padded

// Workspace layout (floats)
#define OFF_MEAN 0
#define OFF_XCT  (OFF_MEAN + 512)                 // 512 x 1232 centered, transposed
#define OFF_COV  (OFF_XCT  + CC*PPAD)             // 512 x 512 cov -> L in place
#define OFF_SCAL (OFF_COV  + CC*CC)               // [0]=logdet [1]=norm
#define OFF_PROB (OFF_SCAL + 16)                  // 1232
#define OFF_WT   (OFF_PROB + PPAD)                // 512 x 1280 weighted feat (c-major)
#define OFF_QT   (OFF_WT   + CC*PW)               // 512 x 320 query feat (c-major)
#define OFF_QQ   (OFF_QT   + CC*QW)               // 128 x 320
#define OFF_QV   (OFF_QQ   + DK*QW)               // 128 x 320
#define OFF_SK   (OFF_QV   + DK*QW)               // 128 x 1280
#define OFF_SV   (OFF_SK   + DK*PW)               // 128 x 1280
#define OFF_SIM  (OFF_SV   + DK*PW)               // 25 x 64 x 64
#define OFF_OBUF (OFF_SIM  + NK*64*64)            // 25 x 128 x 49

__device__ __forceinline__ v8f wmma4(v2f a, v2f b, v8f c) {
  // D = A(16x4,f32) * B(4x16,f32) + C(16x16,f32)
  return __builtin_amdgcn_wmma_f32_16x16x4_f32(
      false, a, false, b, (short)0, c, false, false);
}

__device__ __forceinline__ float waveSum(float v) {
  #pragma unroll
  for (int off = 16; off > 0; off >>= 1) v += __shfl_xor(v, off, 32);
  return v;
}
__device__ __forceinline__ float waveMax(float v) {
  #pragma unroll
  for (int off = 16; off > 0; off >>= 1) v = fmaxf(v, __shfl_xor(v, off, 32));
  return v;
}

// ---- stage 1: per-channel mean over 1225 support pixels ----
__global__ void meanK(const float* __restrict__ S, float* __restrict__ mean) {
  int c = blockIdx.x * blockDim.x + threadIdx.x;
  if (c >= CC) return;
  float s = 0.f;
  for (int nk = 0; nk < NK; nk++) {
    const float* p = S + (size_t)(nk * CC + c) * SS;
    for (int i = 0; i < SS; i++) s += p[i];
  }
  mean[c] = s * (1.0f / (float)NP);
}

// ---- stage 2: centered, transposed features xcT[c][p], zero-padded ----
__global__ void xctK(const float* __restrict__ S, const float* __restrict__ mean,
                     float* __restrict__ xcT) {
  int idx = blockIdx.x * blockDim.x + threadIdx.x;  // < 512*1232
  int c = idx / PPAD, p = idx - c * PPAD;
  float v = 0.f;
  if (p < NP) {
    int nk = p / SS, s = p - nk * SS;
    v = S[(size_t)(nk * CC + c) * SS + s] - mean[c];
  }
  xcT[idx] = v;
}

// ---- stage 3: cov = xc^T xc / (n-1) via WMMA f32 16x16x4 ----
// One wave per 16x16 tile; 32x32 tiles; K = 1232 in steps of 4.
__global__ void covK(const float* __restrict__ xcT, float* __restrict__ cov) {
  int gw = (blockIdx.x * blockDim.x + threadIdx.x) >> 5;  // 0..1023
  int lane = threadIdx.x & 31;
  int I = gw >> 5, J = gw & 31;
  int m = lane & 15, kb = (lane >> 4) * 2;
  const float* arow = xcT + (size_t)(I * 16 + m) * PPAD;
  const float* brow = xcT + (size_t)(J * 16 + m) * PPAD;
  v8f acc = {};
  for (int k0 = 0; k0 < PPAD; k0 += 4) {
    v2f a = *(const v2f*)(arow + k0 + kb);   // 8B-aligned (even offsets)
    v2f b = *(const v2f*)(brow + k0 + kb);
    acc = wmma4(a, b, acc);
  }
  const float inv = 1.0f / (float)(NP - 1);
  int N = lane & 15, Mb = (lane >> 4) * 8;
  #pragma unroll
  for (int r = 0; r < 8; r++) {
    int M = r + Mb;
    cov[(size_t)(I * 16 + M) * CC + J * 16 + N] = acc[r] * inv;
  }
}

// ---- stage 4: in-place Cholesky (lower) + logdet; single workgroup ----
__global__ void cholK(float* __restrict__ A, float* __restrict__ scal) {
  const int n = CC, tid = threadIdx.x, nt = blockDim.x;
  for (int j = 0; j < n; j++) {
    if (tid == 0) A[(size_t)j * n + j] = sqrtf(A[(size_t)j * n + j]);
    __syncthreads();
    float d = A[(size_t)j * n + j];
    for (int i = j + 1 + tid; i < n; i += nt) A[(size_t)i * n + j] /= d;
    __syncthreads();
    for (int i = j + 1 + tid; i < n; i += nt) {
      float lij = A[(size_t)i * n + j];
      for (int k = j + 1; k <= i; k++)
        A[(size_t)i * n + k] -= lij * A[(size_t)k * n + j];
    }
    __syncthreads();
  }
  __shared__ float red[256];
  float s = 0.f;
  for (int j = tid; j < n; j += nt) s += logf(A[(size_t)j * n + j]);
  red[tid] = s;
  __syncthreads();
  for (int off = 128; off > 0; off >>= 1) {
    if (tid < off) red[tid] += red[tid + off];
    __syncthreads();
  }
  if (tid == 0) scal[0] = 2.0f * red[0];
}

// ---- stage 5: forward substitution L y = xc_p, maha = |y|^2, probs ----
// One wave per RHS; y cached in LDS (8 waves * 512 * 4B = 16 KB of 320 KB).
__global__ void solveK(const float* __restrict__ L, const float* __restrict__ xcT,
                       const float* __restrict__ scal, float* __restrict__ probs) {
  __shared__ float y[8][CC];
  int lane = threadIdx.x & 31, wv = threadIdx.x >> 5;
  int p = blockIdx.x * 8 + wv;  // < 1232 (pads read zeros, harmless)
  float maha = 0.f;
  for (int i = 0; i < CC; i++) {
    const float* Lr = L + (size_t)i * CC;
    float part = 0.f;
    for (int k = lane; k < i; k += 32) part += Lr[k] * y[wv][k];
    part = waveSum(part);
    float yi = (xcT[(size_t)i * PPAD + p] - part) / Lr[i];
    if (lane == 0) y[wv][i] = yi;
    maha += yi * yi;
    __syncthreads();  // uniform trip count across all 8 waves
  }
  if (lane == 0 && p < NP) {
    const float log2pi = 1.8378770664093453f;
    float logp = -0.5f * ((float)CC * log2pi + scal[0] + maha);
    probs[p] = expf(logp);
  }
}

// ---- stage 6a: L2 norm of probs ----
__global__ void normK(const float* __restrict__ probs, float* __restrict__ scal) {
  __shared__ float red[256];
  int tid = threadIdx.x;
  float s = 0.f;
  for (int p = tid; p < NP; p += 256) { float v = probs[p]; s += v * v; }
  red[tid] = s;
  __syncthreads();
  for (int off = 128; off > 0; off >>= 1) {
    if (tid < off) red[tid] += red[tid + off];
    __syncthreads();
  }
  if (tid == 0) scal[1] = fmaxf(sqrtf(red[0]), 1e-12f);
}

// ---- stage 6b: weighted features wT[c][p] = (xc+mean)*sigmoid(mask) ----
__global__ void weightK(const float* __restrict__ xcT, const float* __restrict__ mean,
                        const float* __restrict__ probs, const float* __restrict__ scal,
                        float* __restrict__ wT) {
  int idx = blockIdx.x * blockDim.x + threadIdx.x;  // < 512*1280
  int c = idx / PW, p = idx - c * PW;
  float v = 0.f;
  if (p < NP) {
    float mask = probs[p] / scal[1];
    float sg = 1.0f / (1.0f + expf(-mask));
    v = (xcT[(size_t)c * PPAD + p] + mean[c]) * sg;
  }
  wT[idx] = v;
}

// ---- stage 6c: query features transposed qT[c][b*49+s], zero-padded ----
__global__ void qtK(const float* __restrict__ Q, float* __restrict__ qT) {
  int idx = blockIdx.x * blockDim.x + threadIdx.x;  // < 512*320
  int c = idx / QW, bp = idx - c * QW;
  float v = 0.f;
  if (bp < QP) {
    int b = bp / SS, s = bp - b * SS;
    v = Q[(size_t)(b * CC + c) * SS + s];
  }
  qT[idx] = v;
}

// ---- stage 7: 1x1-conv projection GEMM: out[o][col] = sum_c W[o][c]X[c][col] ----
__global__ void projK(const float* __restrict__ W, const float* __restrict__ X,
                      int xs, float* __restrict__ out) {
  int gw = (blockIdx.x * blockDim.x + threadIdx.x) >> 5;
  int lane = threadIdx.x & 31;
  int ntn = xs >> 4;
  int I = gw / ntn, J = gw - I * ntn;
  int m = lane & 15, kb = (lane >> 4) * 2;
  const float* arow = W + (size_t)(I * 16 + m) * CC;
  const float* bcol = X + J * 16 + m;
  v8f acc = {};
  for (int k0 = 0; k0 < CC; k0 += 4) {
    v2f a = *(const v2f*)(arow + k0 + kb);
    v2f b;
    b.x = bcol[(size_t)(k0 + kb) * xs];
    b.y = bcol[(size_t)(k0 + kb + 1) * xs];
    acc = wmma4(a, b, acc);
  }
  int N = lane & 15, Mb = (lane >> 4) * 8;
  #pragma unroll
  for (int r = 0; r < 8; r++) {
    int M = r + Mb;
    out[(size_t)(I * 16 + M) * xs + J * 16 + N] = acc[r];
  }
}

// ---- stage 8: sim[bk][sq][j] = scale * sum_o qq[o][b*49+sq] sk[o][bk*49+j] ----
__global__ void simK(const float* __restrict__ qq, const float* __restrict__ sk,
                     float* __restrict__ simbuf) {
  int gw = (blockIdx.x * blockDim.x + threadIdx.x) >> 5;  // < 400
  int lane = threadIdx.x & 31;
  int bk = gw >> 4, t = gw & 15;
  int I = t >> 2, J = t & 3;
  int b = bk / 5;
  int m = lane & 15, kb = (lane >> 4) * 2;
  int qcol = b * SS + I * 16 + m;
  int pcol = bk * SS + J * 16 + m;
  v8f acc = {};
  for (int k0 = 0; k0 < DK; k0 += 4) {
    v2f a, bb;
    a.x  = qq[(size_t)(k0 + kb)     * QW + qcol];
    a.y  = qq[(size_t)(k0 + kb + 1) * QW + qcol];
    bb.x = sk[(size_t)(k0 + kb)     * PW + pcol];
    bb.y = sk[(size_t)(k0 + kb + 1) * PW + pcol];
    acc = wmma4(a, bb, acc);
  }
  const float scale = 0.08838834764831845f;  // 128^-0.5
  int N = lane & 15, Mb = (lane >> 4) * 8;
  float* outp = simbuf + (size_t)bk * 4096;
  #pragma unroll
  for (int r = 0; r < 8; r++) {
    int M = r + Mb;
    outp[(I * 16 + M) * 64 + J * 16 + N] = acc[r] * scale;
  }
}

// ---- stage 9: row softmax over 49 valid cols; zero pad cols ----
__global__ void softmaxK(float* __restrict__ simbuf) {
  int lane = threadIdx.x & 31;
  int row = blockIdx.x * (blockDim.x >> 5) + (threadIdx.x >> 5);
  if (row >= NP) return;
  int bk = row / SS, sq = row - bk * SS;
  float* rp = simbuf + (size_t)bk * 4096 + sq * 64;
  float v0 = (lane < SS) ? rp[lane] : -INFINITY;
  float v1 = (lane + 32 < SS) ? rp[lane + 32] : -INFINITY;
  float mx = waveMax(fmaxf(v0, v1));
  float e0 = (lane < SS) ? expf(v0 - mx) : 0.f;
  float e1 = (lane + 32 < SS) ? expf(v1 - mx) : 0.f;
  float s = waveSum(e0 + e1);
  float inv = 1.0f / s;
  rp[lane] = (lane < SS) ? e0 * inv : 0.f;
  rp[lane + 32] = (lane + 32 < SS) ? e1 * inv : 0.f;
}

// ---- stage 10: out[bk][o][sq] = sum_j attn[sq][j] sv[o][bk*49+j] ----
__global__ void outK(const float* __restrict__ simbuf, const float* __restrict__ sv,
                     float* __restrict__ obuf) {
  int gw = (blockIdx.x * blockDim.x + threadIdx.x) >> 5;  // < 800
  int lane = threadIdx.x & 31;
  int bk = gw >> 5, t = gw & 31;
  int I = t >> 3, J = t & 7;  // I: sq tile (4), J: o tile (8)
  int pbase = bk * SS;
  int m = lane & 15, kb = (lane >> 4) * 2;
  const float* arow = simbuf + (size_t)bk * 4096 + (I * 16 + m) * 64;
  const float* brow = sv + (size_t)(J * 16 + m) * PW + pbase;
  v8f acc = {};
  for (int k0 = 0; k0 < 64; k0 += 4) {
    v2f a = *(const v2f*)(arow + k0 + kb);
    v2f b;
    b.x = brow[k0 + kb];
    b.y = brow[k0 + kb + 1];
    acc = wmma4(a, b, acc);
  }
  int N = lane & 15, Mb = (lane >> 4) * 8;
  #pragma unroll
  for (int r = 0; r < 8; r++) {
    int M = I * 16 + r + Mb;
    if (M < SS) obuf[(size_t)bk * (DK * SS) + (J * 16 + N) * SS + M] = acc[r];
  }
}

// ---- stage 11: -mean squared distance per (b,k) ----
__global__ void finalK(const float* __restrict__ qv, const float* __restrict__ obuf,
                       float* __restrict__ out) {
  __shared__ float red[256];
  int bk = blockIdx.x, tid = threadIdx.x;
  int b = bk / 5;
  float s = 0.f;
  for (int idx = tid; idx < DK * SS; idx += 256) {
    int o = idx / SS, sq = idx - o * SS;
    float d = qv[(size_t)o * QW + b * SS + sq] - obuf[(size_t)bk * (DK * SS) + idx];
    s += d * d;
  }
  red[tid] = s;
  __syncthreads();
  for (int off = 128; off > 0; off >>= 1) {
    if (tid < off) red[tid] += red[tid + off];
    __syncthreads();
  }
  if (tid == 0) out[bk] = -(red[0] / (float)SS);
}

// ---------------------------------------------------------------------------
extern "C" void kernel_launch(void* const* d_in, const int* in_sizes, int n_in,
                              void* d_out, int out_size, void* d_ws, size_t ws_size,
                              hipStream_t stream) {
  (void)in_sizes; (void)n_in; (void)out_size; (void)ws_size;
  const float* Q   = (const float*)d_in[0];  // (5,512,7,7)
  const float* S   = (const float*)d_in[1];  // (25,512,7,7)
  const float* Wqk = (const float*)d_in[2];  // (128,512)
  const float* Wv  = (const float*)d_in[3];  // (128,512)
  float* ws = (float*)d_ws;

  float* mean   = ws + OFF_MEAN;
  float* xcT    = ws + OFF_XCT;
  float* cov    = ws + OFF_COV;
  float* scal   = ws + OFF_SCAL;
  float* probs  = ws + OFF_PROB;
  float* wT     = ws + OFF_WT;
  float* qT     = ws + OFF_QT;
  float* qq     = ws + OFF_QQ;
  float* qv     = ws + OFF_QV;
  float* sk     = ws + OFF_SK;
  float* sv     = ws + OFF_SV;
  float* simbuf = ws + OFF_SIM;
  float* obuf   = ws + OFF_OBUF;

  meanK    <<<2, 256, 0, stream>>>(S, mean);
  xctK     <<<(CC * PPAD) / 256, 256, 0, stream>>>(S, mean, xcT);
  covK     <<<256, 128, 0, stream>>>(xcT, cov);           // 1024 waves, wmma
  cholK    <<<1, 256, 0, stream>>>(cov, scal);
  solveK   <<<154, 256, 0, stream>>>(cov, xcT, scal, probs);
  normK    <<<1, 256, 0, stream>>>(probs, scal);
  weightK  <<<(CC * PW) / 256, 256, 0, stream>>>(xcT, mean, probs, scal, wT);
  qtK      <<<(CC * QW) / 256, 256, 0, stream>>>(Q, qT);
  projK    <<<40, 128, 0, stream>>>(Wqk, qT, QW, qq);     // wmma
  projK    <<<40, 128, 0, stream>>>(Wv,  qT, QW, qv);     // wmma
  projK    <<<160, 128, 0, stream>>>(Wqk, wT, PW, sk);    // wmma
  projK    <<<160, 128, 0, stream>>>(Wv,  wT, PW, sv);    // wmma
  simK     <<<100, 128, 0, stream>>>(qq, sk, simbuf);     // wmma
  softmaxK <<<154, 256, 0, stream>>>(simbuf);
  outK     <<<200, 128, 0, stream>>>(simbuf, sv, obuf);   // wmma
  finalK   <<<25, 256, 0, stream>>>(qv, obuf, (float*)d_out);
}